// MPNN_60198261621121
// MI455X (gfx1250) — compile-verified
//
#include <hip/hip_runtime.h>
#include <math.h>

typedef float v2f __attribute__((ext_vector_type(2)));
typedef float v8f __attribute__((ext_vector_type(8)));

#define PI_OVER_CUT 0.7853981633974483f
#define SC0 0.28209479177387814f
#define SC1 0.4886025119029199f
#define SC2 1.0925484305920792f
#define SC3 0.31539156525252005f
#define SC4 0.5462742152960396f

__device__ __forceinline__ float sigm(float x) { return 1.f / (1.f + expf(-x)); }

// ---------------------------------------------------------------- zero
__global__ void zero_kernel(float* __restrict__ co, int nco,
                            float* __restrict__ head, int nhead) {
  int i = blockIdx.x * blockDim.x + threadIdx.x;
  if (i < nco) co[i] = 0.f;
  if (i < nhead) head[i] = 0.f;
}

// ---------------------------------------------------------------- species MLPs
__device__ __forceinline__ void lnsilu8(float* z) {
  float m = 0.f;
#pragma unroll
  for (int j = 0; j < 8; ++j) m += z[j];
  m *= 0.125f;
  float v = 0.f;
#pragma unroll
  for (int j = 0; j < 8; ++j) { float t = z[j] - m; v += t * t; }
  v *= 0.125f;
  float inv = rsqrtf(v + 1e-5f);
#pragma unroll
  for (int j = 0; j < 8; ++j) { float y = (z[j] - m) * inv; z[j] = y * sigm(y); }
}

__global__ void coeff_kernel(const float* __restrict__ aspec,
    const float* __restrict__ Wn1, const float* __restrict__ bn1,
    const float* __restrict__ Wn2, const float* __restrict__ bn2,
    const float* __restrict__ Wn3, const float* __restrict__ bn3,
    const float* __restrict__ Wc1, const float* __restrict__ bc1,
    const float* __restrict__ Wc2, const float* __restrict__ bc2,
    const float* __restrict__ Wc3, const float* __restrict__ bc3,
    float* __restrict__ lc, float* __restrict__ nc, int N) {
  int i = blockIdx.x * blockDim.x + threadIdx.x;
  if (i >= N) return;
  float x = aspec[i];
  float z[8], h[8];
  // ---- neigh branch: 1 -> 8 -> 8 -> 24
#pragma unroll
  for (int j = 0; j < 8; ++j) z[j] = x * Wn1[j] + bn1[j];
  lnsilu8(z);
#pragma unroll
  for (int j = 0; j < 8; ++j) {
    float a = bn2[j];
#pragma unroll
    for (int t = 0; t < 8; ++t) a += z[t] * Wn2[t * 8 + j];
    h[j] = a;
  }
  lnsilu8(h);
  for (int o = 0; o < 24; ++o) {
    float a = bn3[o];
#pragma unroll
    for (int t = 0; t < 8; ++t) a += h[t] * Wn3[t * 24 + o];
    nc[(size_t)i * 24 + o] = a;
  }
  // ---- center branch: 1 -> 8 -> 8 -> 64
#pragma unroll
  for (int j = 0; j < 8; ++j) z[j] = x * Wc1[j] + bc1[j];
  lnsilu8(z);
#pragma unroll
  for (int j = 0; j < 8; ++j) {
    float a = bc2[j];
#pragma unroll
    for (int t = 0; t < 8; ++t) a += z[t] * Wc2[t * 8 + j];
    h[j] = a;
  }
  lnsilu8(h);
  for (int o = 0; o < 64; ++o) {
    float a = bc3[o];
#pragma unroll
    for (int t = 0; t < 8; ++t) a += h[t] * Wc3[t * 64 + o];
    lc[(size_t)i * 64 + o] = a;
  }
}

// ---------------------------------------------------------------- edges
__device__ __forceinline__ void sph9(float x, float y, float z, float r2, float* s) {
  s[0] = SC0;
  s[1] = SC1 * y; s[2] = SC1 * z; s[3] = SC1 * x;
  s[4] = SC2 * x * y; s[5] = SC2 * y * z;
  s[6] = SC3 * (3.f * z * z - r2);
  s[7] = SC2 * x * z; s[8] = SC4 * (x * x - y * y);
}

__global__ void edge_fwd(const float* __restrict__ cart,
                         const int* __restrict__ center, const int* __restrict__ neigh,
                         const int* __restrict__ lspec, const float* __restrict__ nc,
                         float* __restrict__ co, int E) {
  int e = blockIdx.x * blockDim.x + threadIdx.x;
  if (e >= E) return;
  int c = center[e], n = neigh[e], ls = lspec[e];
  float dx = cart[n * 3 + 0] - cart[c * 3 + 0];
  float dy = cart[n * 3 + 1] - cart[c * 3 + 1];
  float dz = cart[n * 3 + 2] - cart[c * 3 + 2];
  float r2 = dx * dx + dy * dy + dz * dz;
  float d = sqrtf(r2);
  float cut = 0.5f * cosf(d * PI_OVER_CUT) + 0.5f;
  float cutf = cut * cut;
  float s[9];
  sph9(dx, dy, dz, r2, s);
  const float* ne = nc + (size_t)ls * 24;
  float* dst = co + (size_t)c * 72;
#pragma unroll
  for (int w = 0; w < 8; ++w) {
    float dd = d - ne[16 + w];
    float t = ne[8 + w] * dd;
    float rad = expf(-t * t);
    float aw = cutf * rad * ne[w];
#pragma unroll
    for (int k = 0; k < 9; ++k) atomicAdd(&dst[k * 8 + w], aw * s[k]);
  }
}

__global__ void edge_bwd(const float* __restrict__ cart,
                         const int* __restrict__ center, const int* __restrict__ neigh,
                         const int* __restrict__ lspec, const float* __restrict__ nc,
                         const float* __restrict__ dco, float* __restrict__ force, int E) {
  int e = blockIdx.x * blockDim.x + threadIdx.x;
  if (e >= E) return;
  int c = center[e], n = neigh[e], ls = lspec[e];
  float dx = cart[n * 3 + 0] - cart[c * 3 + 0];
  float dy = cart[n * 3 + 1] - cart[c * 3 + 1];
  float dz = cart[n * 3 + 2] - cart[c * 3 + 2];
  float r2 = dx * dx + dy * dy + dz * dz;
  float d = sqrtf(r2);
  float cosv = cosf(d * PI_OVER_CUT);
  float sinv = sinf(d * PI_OVER_CUT);
  float cut = 0.5f * cosv + 0.5f;
  float cutf = cut * cut;
  float dcut_dd = 2.f * cut * (-0.5f * sinv) * PI_OVER_CUT;
  float s[9];
  sph9(dx, dy, dz, r2, s);
  const float* ne = nc + (size_t)ls * 24;
  const float* g = dco + (size_t)c * 72;
  float gs[9];
#pragma unroll
  for (int k = 0; k < 9; ++k) gs[k] = 0.f;
  float gd = 0.f;
#pragma unroll
  for (int w = 0; w < 8; ++w) {
    float a0 = ne[w], b = ne[8 + w], cen = ne[16 + w];
    float dd = d - cen;
    float t = b * dd;
    float rad = expf(-t * t);
    float Aw = cutf * rad * a0;
    float tw = 0.f;
#pragma unroll
    for (int k = 0; k < 9; ++k) {
      float gkw = g[k * 8 + w];
      tw += gkw * s[k];
      gs[k] += gkw * Aw;
    }
    gd += dcut_dd * (rad * a0 * tw);                 // through cutf
    gd += cutf * a0 * tw * (rad * (-2.f * b * b * dd)); // through rad
  }
  // spherical-harmonic derivatives (s[0] is constant)
  float gx = SC1 * gs[3] + SC2 * dy * gs[4] + (-2.f * SC3 * dx) * gs[6]
           + SC2 * dz * gs[7] + 2.f * SC4 * dx * gs[8];
  float gy = SC1 * gs[1] + SC2 * dx * gs[4] + SC2 * dz * gs[5]
           + (-2.f * SC3 * dy) * gs[6] + (-2.f * SC4 * dy) * gs[8];
  float gz = SC1 * gs[2] + SC2 * dy * gs[5] + 4.f * SC3 * dz * gs[6] + SC2 * dx * gs[7];
  float invd = (r2 > 0.f) ? rsqrtf(r2) : 0.f;
  gx += gd * dx * invd; gy += gd * dy * invd; gz += gd * dz * invd;
  // force = -grad; grad[n] += g, grad[c] -= g
  atomicAdd(&force[n * 3 + 0], -gx);
  atomicAdd(&force[n * 3 + 1], -gy);
  atomicAdd(&force[n * 3 + 2], -gz);
  atomicAdd(&force[c * 3 + 0], gx);
  atomicAdd(&force[c * 3 + 1], gy);
  atomicAdd(&force[c * 3 + 2], gz);
}

// ---------------------------------------------------------------- per-atom WMMA MLP
// 16x64 @ 64x64 FP32 GEMM using V_WMMA_F32_16X16X4_F32.
// A frag (16x4 f32): vgpr r, lane-half kh -> K = 2*kh + r, row M = lane&15
// B frag (4x16 f32): vgpr r, lane-half kh -> K = 2*kh + r, col N = lane&15
// D (16x16 f32):     vgpr r, lane-half kh -> row = r + 8*kh, col = lane&15
__device__ __forceinline__ void gemm16x64(const float* sAmat, const float* __restrict__ W,
                                          bool transp, float* sOut, int tid) {
  const int lane = tid & 31;
  const int wv = tid >> 5;           // 2 waves per block
  const int mn = lane & 15;
  const int kh = lane >> 4;
  for (int nt = 0; nt < 2; ++nt) {
    const int n0 = wv * 32 + nt * 16;
    const int ncol = n0 + mn;
    v8f acc = {0.f, 0.f, 0.f, 0.f, 0.f, 0.f, 0.f, 0.f};
#pragma unroll
    for (int k0 = 0; k0 < 64; k0 += 4) {
      const int ka = k0 + 2 * kh;
      v2f a, b;
      a[0] = sAmat[mn * 64 + ka];
      a[1] = sAmat[mn * 64 + ka + 1];
      if (transp) { b[0] = W[ncol * 64 + ka];  b[1] = W[ncol * 64 + ka + 1]; }
      else        { b[0] = W[ka * 64 + ncol]; b[1] = W[(ka + 1) * 64 + ncol]; }
      acc = __builtin_amdgcn_wmma_f32_16x16x4_f32(false, a, false, b,
                                                  (short)0, acc, false, false);
    }
#pragma unroll
    for (int r = 0; r < 8; ++r) sOut[(r + 8 * kh) * 64 + ncol] = acc[r];
  }
}

__global__ __launch_bounds__(64) void atom_kernel(
    const float* __restrict__ co, const float* __restrict__ cc,
    const float* __restrict__ lc,
    const float* __restrict__ Wo1, const float* __restrict__ bo1,
    const float* __restrict__ Wo2, const float* __restrict__ bo2,
    const float* __restrict__ Wo3, const float* __restrict__ bo3,
    float* __restrict__ dco, float* __restrict__ peratom,
    float* __restrict__ energy, int N) {
  __shared__ float sB[1536];        // cc[0][0..2][8][64]
  __shared__ float sCON[16 * 9 * 64];
  __shared__ float sCO[16 * 72];    // co tile, later reused as lc tile (16*64)
  __shared__ float sA[1024];
  __shared__ float sG[1024];
  __shared__ float sY1[1024];
  __shared__ float sY2[1024];
  __shared__ float sInv1[16], sInv2[16], sM1[16], sM2[16];

  const int tid = threadIdx.x;
  const int base = blockIdx.x * 16;

  for (int idx = tid; idx < 1536; idx += 64) sB[idx] = cc[idx];
  for (int idx = tid; idx < 16 * 72; idx += 64) {
    int r = idx / 72, q = idx % 72, i = base + r;
    sCO[idx] = (i < N) ? co[(size_t)i * 72 + q] : 0.f;
  }
  __syncthreads();

  // con[i,k,m] = sum_w co[i,k,w] * cc0[idxl(k), w, m]
  for (int idx = tid; idx < 16 * 9 * 64; idx += 64) {
    int r = idx / 576, rem = idx % 576;
    int k = rem >> 6, m = rem & 63;
    int jl = (k == 0) ? 0 : ((k < 4) ? 1 : 2);
    const float* bp = &sB[jl * 512 + m];
    const float* cp = &sCO[r * 72 + k * 8];
    float acc = 0.f;
#pragma unroll
    for (int w = 0; w < 8; ++w) acc += cp[w] * bp[w * 64];
    sCON[idx] = acc;
  }
  __syncthreads();

  // reuse sCO for local_coeff tile
  for (int idx = tid; idx < 1024; idx += 64) {
    int r = idx >> 6, i = base + r;
    sCO[idx] = (i < N) ? lc[(size_t)i * 64 + (idx & 63)] : 0.f;
  }
  __syncthreads();

  // dens[i,m] = lc[i,m] * sum_k con^2
  for (int idx = tid; idx < 1024; idx += 64) {
    int r = idx >> 6, m = idx & 63;
    float acc = 0.f;
#pragma unroll
    for (int k = 0; k < 9; ++k) { float v = sCON[r * 576 + k * 64 + m]; acc += v * v; }
    sA[idx] = acc * sCO[idx];
  }
  __syncthreads();

  // ---------------- layer 1 forward
  gemm16x64(sA, Wo1, false, sY1, tid);
  __syncthreads();
  for (int idx = tid; idx < 1024; idx += 64) sY1[idx] += bo1[idx & 63];
  __syncthreads();
  if (tid < 16) {
    float m = 0.f;
    for (int j = 0; j < 64; ++j) m += sY1[tid * 64 + j];
    m *= (1.f / 64.f);
    float v = 0.f;
    for (int j = 0; j < 64; ++j) { float t = sY1[tid * 64 + j] - m; v += t * t; }
    sM1[tid] = m; sInv1[tid] = rsqrtf(v * (1.f / 64.f) + 1e-5f);
  }
  __syncthreads();
  for (int idx = tid; idx < 1024; idx += 64) {
    int r = idx >> 6;
    float y = (sY1[idx] - sM1[r]) * sInv1[r];
    sY1[idx] = y;
    sA[idx] = y * sigm(y);   // h1
  }
  __syncthreads();

  // ---------------- layer 2 forward
  gemm16x64(sA, Wo2, false, sY2, tid);
  __syncthreads();
  for (int idx = tid; idx < 1024; idx += 64) sY2[idx] += bo2[idx & 63];
  __syncthreads();
  if (tid < 16) {
    float m = 0.f;
    for (int j = 0; j < 64; ++j) m += sY2[tid * 64 + j];
    m *= (1.f / 64.f);
    float v = 0.f;
    for (int j = 0; j < 64; ++j) { float t = sY2[tid * 64 + j] - m; v += t * t; }
    sM1[tid] = m; sInv2[tid] = rsqrtf(v * (1.f / 64.f) + 1e-5f);
  }
  __syncthreads();
  for (int idx = tid; idx < 1024; idx += 64) {
    int r = idx >> 6;
    sY2[idx] = (sY2[idx] - sM1[r]) * sInv2[r];
  }
  __syncthreads();

  // ---------------- output layer + energy
  if (tid < 16) {
    int i = base + tid;
    if (i < N) {
      float o = bo3[0];
      for (int m = 0; m < 64; ++m) {
        float y = sY2[tid * 64 + m];
        o += y * sigm(y) * Wo3[m];
      }
      peratom[i] = o;
      atomicAdd(energy, o);
    }
  }

  // ---------------- backward: g_h2 = Wo3; through SiLU(y2)
  for (int idx = tid; idx < 1024; idx += 64) {
    float y = sY2[idx];
    float sg = sigm(y);
    sG[idx] = Wo3[idx & 63] * sg * (1.f + y * (1.f - sg));
  }
  __syncthreads();
  if (tid < 16) {
    float m1 = 0.f, m2 = 0.f;
    for (int j = 0; j < 64; ++j) {
      float gv = sG[tid * 64 + j];
      m1 += gv;
      m2 += gv * sY2[tid * 64 + j];
    }
    sM1[tid] = m1 * (1.f / 64.f); sM2[tid] = m2 * (1.f / 64.f);
  }
  __syncthreads();
  for (int idx = tid; idx < 1024; idx += 64) {
    int r = idx >> 6;
    sG[idx] = sInv2[r] * (sG[idx] - sM1[r] - sY2[idx] * sM2[r]);  // g_z2
  }
  __syncthreads();

  gemm16x64(sG, Wo2, true, sA, tid);   // g_h1 = g_z2 @ W2^T
  __syncthreads();
  for (int idx = tid; idx < 1024; idx += 64) {
    float y = sY1[idx];
    float sg = sigm(y);
    sA[idx] = sA[idx] * sg * (1.f + y * (1.f - sg));  // g_y1
  }
  __syncthreads();
  if (tid < 16) {
    float m1 = 0.f, m2 = 0.f;
    for (int j = 0; j < 64; ++j) {
      float gv = sA[tid * 64 + j];
      m1 += gv;
      m2 += gv * sY1[tid * 64 + j];
    }
    sM1[tid] = m1 * (1.f / 64.f); sM2[tid] = m2 * (1.f / 64.f);
  }
  __syncthreads();
  for (int idx = tid; idx < 1024; idx += 64) {
    int r = idx >> 6;
    sA[idx] = sInv1[r] * (sA[idx] - sM1[r] - sY1[idx] * sM2[r]);  // g_z1
  }
  __syncthreads();

  gemm16x64(sA, Wo1, true, sG, tid);   // g_dens = g_z1 @ W1^T
  __syncthreads();

  // d_co[i,k,w] = sum_m 2*con*lc*g_dens*cc0[idxl(k),w,m]
  for (int idx = tid; idx < 16 * 72; idx += 64) {
    int r = idx / 72, rem = idx % 72;
    int k = rem >> 3, w = rem & 7;
    int jl = (k == 0) ? 0 : ((k < 4) ? 1 : 2);
    const float* bp = &sB[(jl * 8 + w) * 64];
    const float* cn = &sCON[r * 576 + k * 64];
    const float* lr = &sCO[r * 64];
    const float* gr = &sG[r * 64];
    float acc = 0.f;
    for (int m = 0; m < 64; ++m) acc += cn[m] * lr[m] * gr[m] * bp[m];
    int i = base + r;
    if (i < N) dco[(size_t)i * 72 + rem] = 2.f * acc;
  }
}

// ---------------------------------------------------------------- launch
extern "C" void kernel_launch(void* const* d_in, const int* in_sizes, int n_in,
                              void* d_out, int out_size, void* d_ws, size_t ws_size,
                              hipStream_t stream) {
  const float* cart   = (const float*)d_in[0];
  const int*   center = (const int*)d_in[1];
  const int*   neigh  = (const int*)d_in[2];
  const int*   lspec  = (const int*)d_in[3];
  const float* aspec  = (const float*)d_in[6];
  const float* cc     = (const float*)d_in[7];
  const float* Wn1 = (const float*)d_in[8];  const float* bn1 = (const float*)d_in[9];
  const float* Wn2 = (const float*)d_in[10]; const float* bn2 = (const float*)d_in[11];
  const float* Wn3 = (const float*)d_in[12]; const float* bn3 = (const float*)d_in[13];
  const float* Wc1 = (const float*)d_in[14]; const float* bc1 = (const float*)d_in[15];
  const float* Wc2 = (const float*)d_in[16]; const float* bc2 = (const float*)d_in[17];
  const float* Wc3 = (const float*)d_in[18]; const float* bc3 = (const float*)d_in[19];
  const float* Wo1 = (const float*)d_in[20]; const float* bo1 = (const float*)d_in[21];
  const float* Wo2 = (const float*)d_in[22]; const float* bo2 = (const float*)d_in[23];
  const float* Wo3 = (const float*)d_in[24]; const float* bo3 = (const float*)d_in[25];

  const int N = in_sizes[5];   // nlocal
  const int E = in_sizes[1];   // centerlist

  float* out_f   = (float*)d_out;
  float* energy  = out_f;              // [0]
  float* force   = out_f + 1;          // [1 .. 3N]
  float* peratom = out_f + 1 + (size_t)3 * N;

  float* ws  = (float*)d_ws;
  float* lc  = ws;
  float* nc  = lc + (size_t)N * 64;
  float* co  = nc + (size_t)N * 24;
  float* dco = co + (size_t)N * 72;

  int nco = 72 * N;
  int nhead = 1 + 3 * N;
  int nz = (nco > nhead) ? nco : nhead;
  zero_kernel<<<(nz + 255) / 256, 256, 0, stream>>>(co, nco, out_f, nhead);

  coeff_kernel<<<(N + 127) / 128, 128, 0, stream>>>(
      aspec, Wn1, bn1, Wn2, bn2, Wn3, bn3, Wc1, bc1, Wc2, bc2, Wc3, bc3, lc, nc, N);

  edge_fwd<<<(E + 255) / 256, 256, 0, stream>>>(cart, center, neigh, lspec, nc, co, E);

  atom_kernel<<<(N + 15) / 16, 64, 0, stream>>>(
      co, cc, lc, Wo1, bo1, Wo2, bo2, Wo3, bo3, dco, peratom, energy, N);

  edge_bwd<<<(E + 255) / 256, 256, 0, stream>>>(cart, center, neigh, lspec, nc, dco, force, E);
}